// SentenceClassifier_60318520705310
// MI455X (gfx1250) — compile-verified
//
#include <hip/hip_runtime.h>
#include <hip/hip_bf16.h>
#include <math.h>

#define B_ 8
#define L_ 4096
#define H_ 1024
#define S_ 32
#define SEP_ID_ 102
#define NLBL_ 2
#define LN_EPS_ 1e-5f
#define BS_ (B_ * S_)   // 256 segment rows

typedef float v2f __attribute__((ext_vector_type(2)));
typedef float v8f __attribute__((ext_vector_type(8)));

// ---------------------------------------------------------------------------
// k0: per-batch SEP positions (first 32, fill L-1) + per-segment mask counts
// ---------------------------------------------------------------------------
__global__ void k0_sep_counts(const int* __restrict__ ids,
                              const int* __restrict__ mask,
                              int* __restrict__ sep,
                              float* __restrict__ cntf) {
  int b = threadIdx.x;
  if (b >= B_) return;
  const int* idrow = ids + (size_t)b * L_;
  int cnt = 0;
  for (int t = 0; t < L_; ++t) {
    if (idrow[t] == SEP_ID_ && cnt < S_) sep[b * S_ + cnt++] = t;
  }
  for (; cnt < S_; ++cnt) sep[b * S_ + cnt] = L_ - 1;

  const int* mrow = mask + (size_t)b * L_;
  for (int s = 0; s < S_; ++s) {
    int st = (s == 0) ? 0 : sep[b * S_ + s - 1];
    int ed = sep[b * S_ + s];
    int c = 0;
    for (int t = st; t <= ed; ++t) c += mrow[t];
    cntf[b * S_ + s] = (float)(c < 1 ? 1 : c);
  }
}

// ---------------------------------------------------------------------------
// k1: segment means. One block per (b,s), 256 threads, float4 per thread.
//     Streams the 134 MB `seq` tensor exactly once (plus 32 boundary rows).
// ---------------------------------------------------------------------------
__global__ void k1_segmean(const float* __restrict__ seq,
                           const int* __restrict__ mask,
                           const int* __restrict__ sep,
                           const float* __restrict__ cntf,
                           float* __restrict__ reps) {
  int bs = blockIdx.x;          // 0..255
  int b = bs >> 5;
  int s = bs & 31;
  int st = (s == 0) ? 0 : sep[b * S_ + s - 1];
  int ed = sep[b * S_ + s];     // inclusive
  int tid = threadIdx.x;        // 0..255 -> 4 floats each (H=1024)

  const float* base = seq + ((size_t)b * L_ + st) * H_ + tid * 4;
  const int* mrow = mask + (size_t)b * L_;
  float4 acc = make_float4(0.f, 0.f, 0.f, 0.f);
  for (int t = st; t <= ed; ++t) {
    __builtin_prefetch(base + H_, 0, 0);       // global_prefetch_b8
    float4 v = *(const float4*)base;
    float m = (float)mrow[t];
    acc.x += m * v.x; acc.y += m * v.y; acc.z += m * v.z; acc.w += m * v.w;
    base += H_;
  }
  float inv = 1.0f / cntf[bs];
  float4 r = make_float4(acc.x * inv, acc.y * inv, acc.z * inv, acc.w * inv);
  *(float4*)(reps + (size_t)bs * H_ + tid * 4) = r;
}

// ---------------------------------------------------------------------------
// k2: H1 = reps(256x1024) @ W1(1024x1024) + b1 via V_WMMA_F32_16X16X4_F32.
//     Grid (16 m-tiles, 8 n-groups), 8 waves/block -> one 16x16 tile per wave.
//     A layout: lane M = lane&15, VGPR{0,1} hold K = khalf, khalf+1 where
//     khalf = (lane>>4)*2.  B mirrored with lane N = lane&15.
//     C/D: VGPR j holds M = j + 8*(lane>>4), N = lane&15.
// ---------------------------------------------------------------------------
__global__ void k2_gemm1_wmma(const float* __restrict__ Xr,
                              const float* __restrict__ W1,
                              const float* __restrict__ b1,
                              float* __restrict__ H1) {
  int lane = threadIdx.x & 31;
  int wave = threadIdx.x >> 5;
  int m0 = blockIdx.x * 16;
  int n0 = (blockIdx.y * 8 + wave) * 16;
  int row = m0 + (lane & 15);
  int col = n0 + (lane & 15);
  int khalf = (lane >> 4) * 2;                 // 0 or 2

  const float* aptr = Xr + (size_t)row * H_ + khalf;
  const float* bptr = W1 + (size_t)khalf * H_ + col;

  v8f c = {};
#pragma unroll 4
  for (int k = 0; k < H_; k += 4) {
    v2f a = *(const v2f*)aptr;                 // K = khalf, khalf+1 (8B aligned)
    v2f bv;
    bv.x = bptr[0];                            // B[khalf   , col]
    bv.y = bptr[H_];                           // B[khalf+1 , col]
    c = __builtin_amdgcn_wmma_f32_16x16x4_f32(
        /*neg_a=*/false, a, /*neg_b=*/false, bv,
        /*c_mod=*/(short)0, c, /*reuse_a=*/false, /*reuse_b=*/false);
    aptr += 4;
    bptr += (size_t)4 * H_;
  }

  float bias = b1[col];
  int baseRow = m0 + (lane >> 4) * 8;
#pragma unroll
  for (int j = 0; j < 8; ++j) {
    H1[(size_t)(baseRow + j) * H_ + col] = c[j] + bias;
  }
}

// ---------------------------------------------------------------------------
// k3: LayerNorm + exact GELU + (1024 -> 2) projection. One block per row.
// ---------------------------------------------------------------------------
__global__ void k3_ln_gelu_head(const float* __restrict__ H1,
                                const float* __restrict__ ln_g,
                                const float* __restrict__ ln_b,
                                const float* __restrict__ W2,
                                const float* __restrict__ b2,
                                float* __restrict__ out) {
  __shared__ float red[256];
  int row = blockIdx.x;
  int tid = threadIdx.x;

  float4 x = *(const float4*)(H1 + (size_t)row * H_ + tid * 4);
  float ps = x.x + x.y + x.z + x.w;
  float pq = x.x * x.x + x.y * x.y + x.z * x.z + x.w * x.w;

  // mean
  red[tid] = ps; __syncthreads();
  for (int off = 128; off > 0; off >>= 1) {
    if (tid < off) red[tid] += red[tid + off];
    __syncthreads();
  }
  float mu = red[0] * (1.0f / H_);
  __syncthreads();

  // variance (E[x^2] - mu^2)
  red[tid] = pq; __syncthreads();
  for (int off = 128; off > 0; off >>= 1) {
    if (tid < off) red[tid] += red[tid + off];
    __syncthreads();
  }
  float var = red[0] * (1.0f / H_) - mu * mu;
  __syncthreads();
  float rs = rsqrtf(var + LN_EPS_);

  // normalize + gelu + partial dot with W2 columns
  float xs[4] = {x.x, x.y, x.z, x.w};
  float acc0 = 0.f, acc1 = 0.f;
#pragma unroll
  for (int j = 0; j < 4; ++j) {
    int idx = tid * 4 + j;
    float y = (xs[j] - mu) * rs * ln_g[idx] + ln_b[idx];
    float g = 0.5f * y * (1.0f + erff(y * 0.70710678118654752f));
    acc0 += g * W2[idx * NLBL_ + 0];
    acc1 += g * W2[idx * NLBL_ + 1];
  }

  red[tid] = acc0; __syncthreads();
  for (int off = 128; off > 0; off >>= 1) {
    if (tid < off) red[tid] += red[tid + off];
    __syncthreads();
  }
  float o0 = red[0];
  __syncthreads();

  red[tid] = acc1; __syncthreads();
  for (int off = 128; off > 0; off >>= 1) {
    if (tid < off) red[tid] += red[tid + off];
    __syncthreads();
  }
  if (tid == 0) {
    out[row * NLBL_ + 0] = o0 + b2[0];
    out[row * NLBL_ + 1] = red[0] + b2[1];
  }
}

// ---------------------------------------------------------------------------
extern "C" void kernel_launch(void* const* d_in, const int* in_sizes, int n_in,
                              void* d_out, int out_size, void* d_ws, size_t ws_size,
                              hipStream_t stream) {
  const float* seq  = (const float*)d_in[0];  // (B,L,H) fp32
  const int*   ids  = (const int*)d_in[1];    // (B,L)
  const int*   mask = (const int*)d_in[2];    // (B,L)
  const float* W1   = (const float*)d_in[3];  // (H,H)
  const float* b1   = (const float*)d_in[4];  // (H,)
  const float* ln_g = (const float*)d_in[5];  // (H,)
  const float* ln_b = (const float*)d_in[6];  // (H,)
  const float* W2   = (const float*)d_in[7];  // (H,2)
  const float* b2   = (const float*)d_in[8];  // (2,)
  float* out = (float*)d_out;                 // (B,S,2)

  // workspace layout
  char* ws = (char*)d_ws;
  int*   sep  = (int*)ws;                                   // 256 ints
  float* cntf = (float*)(ws + 1024);                        // 256 floats
  float* reps = (float*)(ws + 4096);                        // 256*1024 fp32 (1 MB)
  float* h1   = (float*)(ws + 4096 + (size_t)BS_ * H_ * 4); // 256*1024 fp32 (1 MB)

  k0_sep_counts<<<1, 32, 0, stream>>>(ids, mask, sep, cntf);
  k1_segmean<<<BS_, 256, 0, stream>>>(seq, mask, sep, cntf, reps);
  k2_gemm1_wmma<<<dim3(BS_ / 16, (H_ / 16) / 8), 256, 0, stream>>>(reps, W1, b1, h1);
  k3_ln_gelu_head<<<BS_, 256, 0, stream>>>(h1, ln_g, ln_b, W2, b2, out);
}